// ConcatAttention_52080773431569
// MI455X (gfx1250) — compile-verified
//
#include <hip/hip_runtime.h>
#include <math.h>

// Problem constants from the reference
constexpr int Bsz = 8;
constexpr int S   = 2048;
constexpr int T   = 128;
constexpr int D   = 512;   // DS == DT
constexpr int H   = 128;

typedef float v2f __attribute__((ext_vector_type(2)));
typedef float v4f __attribute__((ext_vector_type(4)));
typedef float v8f __attribute__((ext_vector_type(8)));

__device__ __forceinline__ float fast_tanh(float x) {
#if __has_builtin(__builtin_amdgcn_tanhf)
  return __builtin_amdgcn_tanhf(x);          // gfx1250 v_tanh_f32 (1 trans op)
#else
  // tanh(x) = 1 - 2/(e^{2x}+1); e^{2x} = exp2(2*log2(e)*x).  2 trans ops.
  float e = __builtin_amdgcn_exp2f(x * 2.8853900817779268f);
  return 1.0f - 2.0f * __builtin_amdgcn_rcpf(e + 1.0f);
#endif
}

// ---------------------------------------------------------------------------
// Projection GEMM: P[M x 128] = X[M x 512] * Wp[512 x 128]   (f32 WMMA)
// Block = 256 threads (8 waves) -> 16 rows x 128 cols.
// Wave w owns the 16-wide column tile n0 = 16*w; 128 x V_WMMA_F32_16X16X4_F32.
// A tile staged in LDS with row stride 516 floats: fragment-read bank =
// (4*L + k + khalf) mod 64 -> conflict free for all 32 lanes (dual b64 loads).
// ---------------------------------------------------------------------------
#define A_STRIDE 516

__global__ __launch_bounds__(256)
void proj_kernel(const float* __restrict__ X, const float* __restrict__ Wp,
                 float* __restrict__ P) {
  __shared__ float atile[16 * A_STRIDE];

  const int tid  = threadIdx.x;
  const int lane = tid & 31;
  const int wave = tid >> 5;             // 0..7 -> N tile
  const int row0 = blockIdx.x * 16;

  // Cooperative fill: 16 rows x 512 f32 = 2048 float4, 8 per thread (coalesced)
  #pragma unroll
  for (int j = 0; j < 8; ++j) {
    int idx4 = tid + j * 256;            // 0..2047
    int r    = idx4 >> 7;                // /128
    int c4   = (idx4 & 127) << 2;
    v4f val  = *(const v4f*)(X + (size_t)(row0 + r) * D + c4);
    *(v4f*)(atile + r * A_STRIDE + c4) = val;   // 516%4==0 -> 16B aligned
  }
  __syncthreads();

  const int col   = wave * 16 + (lane & 15);
  const int khalf = (lane >> 4) * 2;     // 0 (lanes 0-15) or 2 (lanes 16-31)
  const float* arow = atile + (lane & 15) * A_STRIDE + khalf;

  v8f acc = {};
  for (int k = 0; k < D; k += 4) {
    // A 16x4 frag: lanes 0-15 -> {K=k,k+1}; lanes 16-31 -> {K=k+2,k+3}
    v2f a = *(const v2f*)(arow + k);
    // B 4x16 frag: lanes 0-15 -> rows k,k+1; lanes 16-31 -> rows k+2,k+3
    v2f b;
    b.x = Wp[(size_t)(k + khalf) * H + col];
    b.y = Wp[(size_t)(k + khalf + 1) * H + col];
    acc = __builtin_amdgcn_wmma_f32_16x16x4_f32(
        /*neg_a=*/false, a, /*neg_b=*/false, b,
        /*c_mod=*/(short)0, acc, /*reuse_a=*/false, /*reuse_b=*/false);
  }

  // D 16x16 f32: VGPR r -> M=r (lanes 0-15) / M=r+8 (lanes 16-31), N=lane&15
  const int rbase = row0 + (lane >> 4) * 8;
  #pragma unroll
  for (int r = 0; r < 8; ++r)
    P[(size_t)(rbase + r) * H + col] = acc[r];
}

// ---------------------------------------------------------------------------
// Score kernel: out[b,t,s] = mask ? sum_h v[h]*tanh(srcp[b,s,h]+tgtp[b,t,h])
//                                 : -inf
// Block = (b, 64-s chunk), 256 threads (8 waves).
// LDS: sp[64][129] (stride 129 -> per-lane reads conflict free),
//      tp[128][128] (broadcast reads), v[128], masks.
// Wave w: s = (w&1)*32 + lane  (coalesced stores), t range (w>>1)*32..+31,
//         processed as 2 tiles of 16 accumulators (trans-unit bound).
// ---------------------------------------------------------------------------
#define SCHUNK    64
#define SP_STRIDE 129
#define SP_FLOATS (SCHUNK * SP_STRIDE)        // 8256
#define TP_FLOATS (T * H)                     // 16384
#define TTILE     16

__global__ __launch_bounds__(256)
void score_kernel(const float* __restrict__ srcp, const float* __restrict__ tgtp,
                  const int* __restrict__ smask, const int* __restrict__ tmask,
                  const float* __restrict__ v, float* __restrict__ out) {
  extern __shared__ float lds[];
  float* sp = lds;                           // 64 x 129
  float* tp = lds + SP_FLOATS;               // 128 x 128
  float* vv = tp + TP_FLOATS;                // 128
  int*   sm = (int*)(vv + H);                // 64
  int*   tm = sm + SCHUNK;                   // 128

  const int tid = threadIdx.x;
  const int b   = blockIdx.y;
  const int s0  = blockIdx.x * SCHUNK;

  // src_p chunk: 64 x 128 f32 = 2048 float4, 8 per thread; scalar LDS stores
  // because stride 129 breaks 16B alignment.
  #pragma unroll
  for (int j = 0; j < 8; ++j) {
    int idx4 = tid + j * 256;
    int r = idx4 >> 5, c4 = (idx4 & 31) << 2;
    v4f val = *(const v4f*)(srcp + (size_t)(b * S + s0 + r) * H + c4);
    float* dst = sp + r * SP_STRIDE + c4;
    dst[0] = val.x; dst[1] = val.y; dst[2] = val.z; dst[3] = val.w;
  }
  // tgt_p: 128 x 128 f32 = 4096 float4, 16 per thread (aligned b128 stores)
  #pragma unroll
  for (int j = 0; j < 16; ++j) {
    int idx4 = tid + j * 256;
    int r = idx4 >> 5, c4 = (idx4 & 31) << 2;
    v4f val = *(const v4f*)(tgtp + (size_t)(b * T + r) * H + c4);
    *(v4f*)(tp + r * H + c4) = val;
  }
  if (tid < H)      vv[tid] = v[tid];
  if (tid < SCHUNK) sm[tid] = smask[b * S + s0 + tid];
  if (tid < T)      tm[tid] = tmask[b * T + tid];
  __syncthreads();

  const int lane = tid & 31;
  const int wave = tid >> 5;
  const int s    = (wave & 1) * 32 + lane;    // 0..63, lane-consecutive
  const int t0   = (wave >> 1) * 32;          // 0,32,64,96
  const float* sprow = sp + s * SP_STRIDE;
  const int   smv    = sm[s];
  const float NEG_INF = -__builtin_inff();

  #pragma unroll
  for (int tt = 0; tt < 32 / TTILE; ++tt) {
    const int tb = t0 + tt * TTILE;
    float acc[TTILE];
    #pragma unroll
    for (int e = 0; e < TTILE; ++e) acc[e] = 0.f;

    for (int h = 0; h < H; ++h) {
      float spv = sprow[h];                   // conflict-free (stride 129)
      float vh  = vv[h];                      // broadcast
      #pragma unroll
      for (int e = 0; e < TTILE; ++e) {
        float x = spv + tp[(tb + e) * H + h]; // broadcast
        acc[e] = fmaf(fast_tanh(x), vh, acc[e]);
      }
    }
    #pragma unroll
    for (int e = 0; e < TTILE; ++e) {
      int t = tb + e;
      float val = (smv != 0 && tm[t] != 0) ? acc[e] : NEG_INF;
      out[(size_t)(b * T + t) * S + s0 + s] = val;   // lane-coalesced
    }
  }
}

// ---------------------------------------------------------------------------
extern "C" void kernel_launch(void* const* d_in, const int* in_sizes, int n_in,
                              void* d_out, int out_size, void* d_ws, size_t ws_size,
                              hipStream_t stream) {
  (void)in_sizes; (void)n_in; (void)out_size; (void)ws_size;
  const float* src   = (const float*)d_in[0];  // (8,2048,512) f32
  const float* tgt   = (const float*)d_in[1];  // (8,128,512)  f32
  const int*   smask = (const int*)d_in[2];    // (8,2048)     i32
  const int*   tmask = (const int*)d_in[3];    // (8,128)      i32
  const float* W     = (const float*)d_in[4];  // (1024,128)   f32
  const float* v     = (const float*)d_in[5];  // (128,)       f32
  float* out  = (float*)d_out;                 // (8,128,2048) f32

  float* srcp = (float*)d_ws;                  // (B*S) x H = 8 MB
  float* tgtp = srcp + (size_t)Bsz * S * H;    // (B*T) x H = 0.5 MB

  // src projection: M = B*S = 16384 rows
  proj_kernel<<<dim3((Bsz * S) / 16), 256, 0, stream>>>(src, W, srcp);
  // tgt projection: M = B*T = 1024 rows, uses W rows [512, 1024)
  proj_kernel<<<dim3((Bsz * T) / 16), 256, 0, stream>>>(tgt, W + (size_t)D * H, tgtp);

  size_t shmem = (size_t)(SP_FLOATS + TP_FLOATS + H) * sizeof(float)
               + (size_t)(SCHUNK + T) * sizeof(int);
  score_kernel<<<dim3(S / SCHUNK, Bsz), 256, shmem, stream>>>(
      srcp, tgtp, smask, tmask, v, out);
}